// Model_41059887350378
// MI455X (gfx1250) — compile-verified
//
#include <hip/hip_runtime.h>
#include <cstdint>

// ---------------------------------------------------------------------------
// 2-layer GCN for MI455X (gfx1250, wave32).
//  x:[N,512] fp32, edge_index:[2,E] int32, W1:[512,16], b1:[16], W2:[16,40], b2:[40]
//  out:[N,40] fp32
// GEMMs use v_wmma_f32_16x16x4_f32 (fp32 matrix path). Aggregation uses
// global_atomic_add_f32 into L2-resident buffers (all feature maps << 192MB L2).
// ---------------------------------------------------------------------------

typedef __attribute__((ext_vector_type(2))) float v2f;
typedef __attribute__((ext_vector_type(8))) float v8f;

#define F_IN  512
#define HID   16
#define OUT_F 40

// ---------------- utility -------------------------------------------------
__global__ void fill_f32(float* __restrict__ p, float v, long long n) {
  long long i = (long long)blockIdx.x * blockDim.x + threadIdx.x;
  if (i < n) p[i] = v;
}

// deg[i] starts at 1.0 (self-loop); add 1 per incident target-edge
__global__ void deg_count(const int* __restrict__ row, float* deg, int E) {
  int e = blockIdx.x * blockDim.x + threadIdx.x;
  if (e < E) atomicAdd(&deg[row[e]], 1.0f);
}

__global__ void deg_to_dinv(float* __restrict__ d, int N) {
  int i = blockIdx.x * blockDim.x + threadIdx.x;
  if (i < N) {
    float v = d[i];
    d[i] = (v > 0.0f) ? rsqrtf(v) : 0.0f;
  }
}

// ---------------- layer 1 GEMM: h = x @ W1  ([N,512]x[512,16]) ------------
// One wave per 16-row tile. W1 (32KB) staged in LDS once per block.
// 128 k-steps of v_wmma_f32_16x16x4_f32; x rows streamed from HBM w/ prefetch.
__global__ void __launch_bounds__(256)
gemm_x_w1(const float* __restrict__ x, const float* __restrict__ W1,
          float* __restrict__ h, int N) {
  __shared__ float sW[F_IN * HID];                 // 32 KB of 320 KB/WGP
  for (int i = threadIdx.x; i < F_IN * HID; i += 256) sW[i] = W1[i];
  __syncthreads();

  int wave = threadIdx.x >> 5;                     // 8 waves / block
  int lane = threadIdx.x & 31;
  int row0 = (blockIdx.x * 8 + wave) * 16;
  if (row0 >= N) return;                           // whole-wave exit (EXEC stays all-1)

  int m  = lane & 15;                              // M for A, N for B (same lane slot)
  int kh = lane >> 4;                              // K-half select (ISA 32-bit A layout)
  const float* xrow = x + (size_t)(row0 + m) * F_IN;

  v8f acc = {};
  for (int k0 = 0; k0 < F_IN; k0 += 32) {
    __builtin_prefetch(xrow + k0 + 64, 0, 3);      // global_prefetch_b8, stream x ahead
#pragma unroll
    for (int kk = 0; kk < 32; kk += 4) {
      int k = k0 + kk;
      v2f a, b;
      a.x = xrow[k + 2 * kh + 0];                  // A: (M=m, K=2*kh)   in VGPR0
      a.y = xrow[k + 2 * kh + 1];                  //    (M=m, K=2*kh+1) in VGPR1
      b.x = sW[(k + 2 * kh + 0) * HID + m];        // B: (K, N=m) mirrored layout
      b.y = sW[(k + 2 * kh + 1) * HID + m];
      acc = __builtin_amdgcn_wmma_f32_16x16x4_f32(
          false, a, false, b, (short)0, acc, false, false);
    }
  }
  // C/D layout: VGPR g -> M = g + 8*kh, N = m
  float* hrow = h + (size_t)row0 * HID;
#pragma unroll
  for (int g = 0; g < 8; ++g)
    hrow[(kh * 8 + g) * HID + m] = acc[g];
}

// ---------------- edge aggregation, 16 features ---------------------------
// thread t -> (edge e = t>>4, feature f = t&15): coalesced gather of h[col],
// scattered atomic add into L2-resident out[row].
__global__ void agg16(const float* __restrict__ h, const int* __restrict__ row,
                      const int* __restrict__ col, const float* __restrict__ dinv,
                      float* out, long long total) {
  long long t = (long long)blockIdx.x * blockDim.x + threadIdx.x;
  if (t >= total) return;
  int e = (int)(t >> 4);
  int f = (int)(t & 15);
  int r = row[e], c = col[e];
  float nrm = dinv[r] * dinv[c];
  atomicAdd(out + (size_t)r * HID + f, h[(size_t)c * HID + f] * nrm);
}

// self-loop (norm = dinv^2) + bias + ReLU, in place (agg -> h1)
__global__ void finish1(float* __restrict__ agg, const float* __restrict__ hlin,
                        const float* __restrict__ dinv, const float* __restrict__ b1,
                        long long total) {
  long long t = (long long)blockIdx.x * blockDim.x + threadIdx.x;
  if (t >= total) return;
  int i = (int)(t >> 4);
  int f = (int)(t & 15);
  float di = dinv[i];
  float v  = agg[t] + hlin[t] * di * di + b1[f];
  agg[t] = fmaxf(v, 0.0f);
}

// ---------------- layer 2 GEMM: h2 = h1 @ W2  ([N,16]x[16,40]) ------------
// W2 zero-padded to 16x48 in LDS; one wave -> 16x48 tile via 3 accumulators
// sharing each A fragment. Only cols < 40 stored.
__global__ void __launch_bounds__(256)
gemm_h1_w2(const float* __restrict__ h1, const float* __restrict__ W2,
           float* __restrict__ h2, int N) {
  __shared__ float sW[HID * 48];
  for (int i = threadIdx.x; i < HID * 48; i += 256) {
    int r = i / 48, c = i % 48;
    sW[i] = (c < OUT_F) ? W2[r * OUT_F + c] : 0.0f;
  }
  __syncthreads();

  int wave = threadIdx.x >> 5;
  int lane = threadIdx.x & 31;
  int row0 = (blockIdx.x * 8 + wave) * 16;
  if (row0 >= N) return;

  int m  = lane & 15;
  int kh = lane >> 4;
  const float* arow = h1 + (size_t)(row0 + m) * HID;

  v8f acc0 = {}, acc1 = {}, acc2 = {};
#pragma unroll
  for (int k = 0; k < HID; k += 4) {
    v2f a;
    a.x = arow[k + 2 * kh + 0];
    a.y = arow[k + 2 * kh + 1];
    v2f b0, b1v, b2v;
    b0.x  = sW[(k + 2 * kh + 0) * 48 + m];       b0.y  = sW[(k + 2 * kh + 1) * 48 + m];
    b1v.x = sW[(k + 2 * kh + 0) * 48 + 16 + m];  b1v.y = sW[(k + 2 * kh + 1) * 48 + 16 + m];
    b2v.x = sW[(k + 2 * kh + 0) * 48 + 32 + m];  b2v.y = sW[(k + 2 * kh + 1) * 48 + 32 + m];
    acc0 = __builtin_amdgcn_wmma_f32_16x16x4_f32(false, a, false, b0,  (short)0, acc0, false, false);
    acc1 = __builtin_amdgcn_wmma_f32_16x16x4_f32(false, a, false, b1v, (short)0, acc1, false, false);
    acc2 = __builtin_amdgcn_wmma_f32_16x16x4_f32(false, a, false, b2v, (short)0, acc2, false, false);
  }
#pragma unroll
  for (int g = 0; g < 8; ++g) {
    float* orow = h2 + (size_t)(row0 + kh * 8 + g) * OUT_F;
    orow[m]      = acc0[g];
    orow[16 + m] = acc1[g];
    if (m < 8) orow[32 + m] = acc2[g];             // cols 32..39 only
  }
}

// ---------------- edge aggregation, 40 features ---------------------------
__global__ void agg40(const float* __restrict__ h2, const int* __restrict__ row,
                      const int* __restrict__ col, const float* __restrict__ dinv,
                      float* out, long long total) {
  long long t = (long long)blockIdx.x * blockDim.x + threadIdx.x;
  if (t >= total) return;
  int e = (int)(t / OUT_F);
  int f = (int)(t % OUT_F);
  int r = row[e], c = col[e];
  float nrm = dinv[r] * dinv[c];
  atomicAdd(out + (size_t)r * OUT_F + f, h2[(size_t)c * OUT_F + f] * nrm);
}

// self-loop + bias, in place on d_out
__global__ void finish2(float* __restrict__ out, const float* __restrict__ h2,
                        const float* __restrict__ dinv, const float* __restrict__ b2,
                        long long total) {
  long long t = (long long)blockIdx.x * blockDim.x + threadIdx.x;
  if (t >= total) return;
  int i = (int)(t / OUT_F);
  int f = (int)(t % OUT_F);
  float di = dinv[i];
  out[t] = out[t] + h2[t] * di * di + b2[f];
}

// ---------------------------------------------------------------------------
extern "C" void kernel_launch(void* const* d_in, const int* in_sizes, int n_in,
                              void* d_out, int out_size, void* d_ws, size_t ws_size,
                              hipStream_t stream) {
  const float* x   = (const float*)d_in[0];
  const int*  eidx = (const int*) d_in[1];   // int32 (JAX x64 disabled)
  const float* W1  = (const float*)d_in[2];
  const float* b1  = (const float*)d_in[3];
  const float* W2  = (const float*)d_in[4];
  const float* b2  = (const float*)d_in[5];
  float* out = (float*)d_out;

  const int N = in_sizes[0] / F_IN;    // 100000
  const int E = in_sizes[1] / 2;       // 3200000
  const int* row = eidx;               // edge_index[0]
  const int* col = eidx + E;           // edge_index[1]

  // workspace layout (floats): dinv[N] | hlin[16N] | h1/agg1[16N] | h2[40N]  (~29 MB)
  float* ws   = (float*)d_ws;
  float* dinv = ws;
  float* hlin = dinv + N;
  float* h1   = hlin + (size_t)N * HID;
  float* h2   = h1   + (size_t)N * HID;

  const int B = 256;
  const unsigned gN    = (unsigned)((N + B - 1) / B);
  const unsigned gE    = (unsigned)((E + B - 1) / B);
  const unsigned gTile = (unsigned)(((N + 15) / 16 + 7) / 8);   // 8 waves/block
  const long long n16 = (long long)N * HID;
  const long long n40 = (long long)N * OUT_F;
  const long long e16 = (long long)E * HID;
  const long long e40 = (long long)E * OUT_F;

  // degree / normalization
  fill_f32   <<<gN, B, 0, stream>>>(dinv, 1.0f, N);             // self-loop count
  deg_count  <<<gE, B, 0, stream>>>(row, dinv, E);
  deg_to_dinv<<<gN, B, 0, stream>>>(dinv, N);

  // layer 1
  gemm_x_w1<<<gTile, B, 0, stream>>>(x, W1, hlin, N);
  fill_f32 <<<(unsigned)((n16 + B - 1) / B), B, 0, stream>>>(h1, 0.0f, n16);
  agg16    <<<(unsigned)((e16 + B - 1) / B), B, 0, stream>>>(hlin, row, col, dinv, h1, e16);
  finish1  <<<(unsigned)((n16 + B - 1) / B), B, 0, stream>>>(h1, hlin, dinv, b1, n16);

  // layer 2
  gemm_h1_w2<<<gTile, B, 0, stream>>>(h1, W2, h2, N);
  fill_f32  <<<(unsigned)((n40 + B - 1) / B), B, 0, stream>>>(out, 0.0f, n40);
  agg40     <<<(unsigned)((e40 + B - 1) / B), B, 0, stream>>>(h2, row, col, dinv, out, e40);
  finish2   <<<(unsigned)((n40 + B - 1) / B), B, 0, stream>>>(out, h2, dinv, b2, n40);
}